// GNNModel_8203387535899
// MI455X (gfx1250) — compile-verified
//
#include <hip/hip_runtime.h>
#include <hip/hip_bf16.h>

// ---------------------------------------------------------------------------
// GATv2 GNN forward for MI455X (gfx1250, wave32). All GEMMs via
// V_WMMA_F32_16X16X4_F32 (full fp32; model is bandwidth-bound, so fp32 matrix
// ops are the right precision/throughput point). Edge encoder fuses two GEMMs
// + L2-normalize through LDS; the per-edge transform ea2@We is computed inside
// the attention-score kernel with hoisted A fragments (loaded once, reused
// across all N tiles) and reduced in-register — never spilled to HBM.
// ---------------------------------------------------------------------------

typedef float v2f __attribute__((ext_vector_type(2)));
typedef float v8f __attribute__((ext_vector_type(8)));

__device__ __forceinline__ v8f wmma4(v2f a, v2f b, v8f c) {
  // D = A(16x4) * B(4x16) + C(16x16), fp32, wave32
  return __builtin_amdgcn_wmma_f32_16x16x4_f32(
      /*neg_a=*/false, a, /*neg_b=*/false, b,
      /*c_mod=*/(short)0, c, /*reuse_a=*/false, /*reuse_b=*/false);
}

__device__ __forceinline__ void atomicMaxF(float* addr, float v) {
  // sign-punned monotonic max (valid for mixed signs; init to -large)
  if (v >= 0.0f) atomicMax((int*)addr, __float_as_int(v));
  else           atomicMin((unsigned int*)addr, __float_as_uint(v));
}

// ---------------------------------------------------------------------------
// Generic WMMA GEMM: out[M,N] = epilogue(A[M,K] @ W[K,N] + bias)
// MODE: 0 none, 1 leaky(slope), 2 ELU, 3 skip (0.01*g + add), 4 sigmoid mix
// One wave (32 threads) per 16x16 output tile. M,N multiples of 16, K of 4.
// ---------------------------------------------------------------------------
template<int MODE>
__global__ void wmma_gemm_kernel(const float* __restrict__ A,
                                 const float* __restrict__ W,
                                 const float* __restrict__ bias,
                                 const float* __restrict__ add,
                                 const float* __restrict__ sp,
                                 float slope,
                                 float* __restrict__ out,
                                 int M, int K, int N) {
  const int lane  = threadIdx.x;
  const int ntile = N >> 4;
  const int tile  = blockIdx.x;
  const int mbase = (tile / ntile) << 4;
  const int nbase = (tile % ntile) << 4;
  if (mbase >= M) return;
  const int half = lane >> 4;
  const int l15  = lane & 15;
  const int koff = half << 1;               // fp32 A layout: K = koff, koff+1
  const float* Arow = A + (size_t)(mbase + l15) * K;
  const int colB = nbase + l15;

  v8f acc = {0.f,0.f,0.f,0.f,0.f,0.f,0.f,0.f};
  for (int k = 0; k < K; k += 4) {
    v2f a = *(const v2f*)(Arow + k + koff);           // b64 load
    v2f b;
    b.x = W[(size_t)(k + koff)     * N + colB];
    b.y = W[(size_t)(k + koff + 1) * N + colB];
    acc = wmma4(a, b, acc);
  }

  float s = 0.f;
  if (MODE == 4) s = 1.f / (1.f + __expf(-sp[0]));
  const float bc = bias ? bias[colB] : 0.f;
#pragma unroll
  for (int r = 0; r < 8; ++r) {
    const int row = mbase + r + (half << 3);          // C/D layout
    float g = acc[r] + bc;
    float o;
    if      (MODE == 0) o = g;
    else if (MODE == 1) o = (g > 0.f) ? g : g * slope;
    else if (MODE == 2) o = (g > 0.f) ? g : (__expf(g) - 1.f);
    else if (MODE == 3) o = 0.01f * g + add[(size_t)row * N + colB];
    else                o = s * g + (1.f - s) * add[(size_t)row * N + colB];
    out[(size_t)row * N + colB] = o;
  }
}

// ---------------------------------------------------------------------------
// BatchNorm1d (eval)
// ---------------------------------------------------------------------------
__global__ void bn_kernel(const float* __restrict__ x, const float* __restrict__ g,
                          const float* __restrict__ b, const float* __restrict__ m,
                          const float* __restrict__ v, float* __restrict__ y, int total) {
  int i = blockIdx.x * blockDim.x + threadIdx.x;
  if (i >= total) return;
  int c = i & 63;
  y[i] = (x[i] - m[c]) * rsqrtf(v[c] + 1e-5f) * g[c] + b[c];
}

// ---------------------------------------------------------------------------
// Fused edge encoder: leaky(leaky(attr@W1+b1)@W2+b2), row-L2-normalized.
// One wave per 16 edges. Both GEMMs are WMMA; relayout through LDS.
// A fragments hoisted and reused across all N tiles.
// ---------------------------------------------------------------------------
__global__ void edge_encoder_kernel(const float* __restrict__ EA,
                                    const float* __restrict__ W1, const float* __restrict__ B1,
                                    const float* __restrict__ W2, const float* __restrict__ B2,
                                    float* __restrict__ ea2) {
  __shared__ float ldsH[16 * 128];
  __shared__ float ldsE[16 * 64];
  const int lane = threadIdx.x;
  const int tb   = blockIdx.x;
  const int half = lane >> 4, l15 = lane & 15, koff = half << 1;
  const float* Arow = EA + (size_t)(tb * 16 + l15) * 16;

  // stage 1: [16,16] @ [16,128] -> LDS   (A: 4 fragments, reused 8x)
  v2f av1[4];
#pragma unroll
  for (int kk = 0; kk < 4; ++kk) av1[kk] = *(const v2f*)(Arow + kk * 4 + koff);

  for (int nt = 0; nt < 8; ++nt) {
    const int col = (nt << 4) + l15;
    v8f acc = {0.f,0.f,0.f,0.f,0.f,0.f,0.f,0.f};
#pragma unroll
    for (int kk = 0; kk < 4; ++kk) {
      v2f b;
      b.x = W1[(kk * 4 + koff) * 128 + col];
      b.y = W1[(kk * 4 + koff + 1) * 128 + col];
      acc = wmma4(av1[kk], b, acc);
    }
    const float bc = B1[col];
#pragma unroll
    for (int r = 0; r < 8; ++r) {
      float g = acc[r] + bc;
      ldsH[(r + (half << 3)) * 128 + col] = (g > 0.f) ? g : 0.01f * g;
    }
  }
  __syncthreads();

  // stage 2: [16,128] @ [128,64] -> LDS  (A: 32 fragments from LDS, reused 4x)
  v2f av2[32];
#pragma unroll
  for (int kk = 0; kk < 32; ++kk)
    av2[kk] = *(const v2f*)(&ldsH[l15 * 128 + kk * 4 + koff]);   // ds_load_b64

  for (int nt = 0; nt < 4; ++nt) {
    const int col = (nt << 4) + l15;
    v8f acc = {0.f,0.f,0.f,0.f,0.f,0.f,0.f,0.f};
#pragma unroll
    for (int kk = 0; kk < 32; ++kk) {
      v2f b;
      b.x = W2[(kk * 4 + koff) * 64 + col];
      b.y = W2[(kk * 4 + koff + 1) * 64 + col];
      acc = wmma4(av2[kk], b, acc);
    }
    const float bc = B2[col];
#pragma unroll
    for (int r = 0; r < 8; ++r) {
      float g = acc[r] + bc;
      ldsE[(r + (half << 3)) * 64 + col] = (g > 0.f) ? g : 0.01f * g;
    }
  }
  __syncthreads();

  // stage 3: L2 normalize rows; lanes l15 / l15+16 split cols of row l15
  float ss = 0.f;
#pragma unroll
  for (int j = 0; j < 32; j += 4) {
    float4 v = *(const float4*)&ldsE[l15 * 64 + half * 32 + j];
    ss += v.x * v.x + v.y * v.y + v.z * v.z + v.w * v.w;
  }
  ss += __shfl_xor(ss, 16);
  const float rn = 1.f / fmaxf(sqrtf(ss), 1e-12f);
#pragma unroll
  for (int j = 0; j < 32; j += 4) {
    float4 v = *(const float4*)&ldsE[l15 * 64 + half * 32 + j];
    v.x *= rn; v.y *= rn; v.z *= rn; v.w *= rn;
    *(float4*)&ea2[(size_t)(tb * 16 + l15) * 64 + half * 32 + j] = v;  // b128 store
  }
}

// ---------------------------------------------------------------------------
// Self loops
// ---------------------------------------------------------------------------
__global__ void build_edges_kernel(const int* __restrict__ ei, int* __restrict__ src2,
                                   int* __restrict__ dst2, int E, int N) {
  int i = blockIdx.x * blockDim.x + threadIdx.x;
  if (i >= E + N) return;
  if (i < E) { src2[i] = ei[i]; dst2[i] = ei[E + i]; }
  else       { src2[i] = i - E; dst2[i] = i - E; }
}

__global__ void loop_accum_kernel(float* __restrict__ ea2, const int* __restrict__ dst,
                                  float* __restrict__ deg, int E) {
  int i = blockIdx.x * blockDim.x + threadIdx.x;
  if (i >= E * 64) return;
  int e = i >> 6, c = i & 63;
  int d = dst[e];
  atomicAdd(&ea2[(size_t)(E + d) * 64 + c], ea2[(size_t)e * 64 + c]);
  if (c == 0) atomicAdd(&deg[d], 1.0f);
}

__global__ void loop_div_kernel(float* __restrict__ ea2, const float* __restrict__ deg,
                                int E, int N) {
  int i = blockIdx.x * blockDim.x + threadIdx.x;
  if (i >= N * 64) return;
  int n = i >> 6, c = i & 63;
  ea2[(size_t)(E + n) * 64 + c] /= fmaxf(deg[n], 1.0f);
}

__global__ void fill_kernel(float* __restrict__ p, float v, int n) {
  int i = blockIdx.x * blockDim.x + threadIdx.x;
  if (i < n) p[i] = v;
}

// ---------------------------------------------------------------------------
// GATv2 pass A: per 16-edge tile, compute ee = ea2@We via WMMA (A fragments
// hoisted: one b64 load each, reused across all N tiles), fuse
// m = leaky(xl[src]+xr[dst]+ee, 0.2), score[e,h] = sum_c m*att[h,c].
// Writes scores to sbuf[E2,4] and segment-max into emax via float atomics.
// ---------------------------------------------------------------------------
__global__ void gat_score_kernel(const float* __restrict__ ea2, const float* __restrict__ We,
                                 const float* __restrict__ xl, const float* __restrict__ xr,
                                 const float* __restrict__ att,
                                 const int* __restrict__ src, const int* __restrict__ dst,
                                 float* __restrict__ sbuf, float* __restrict__ emax,
                                 int HC, int C) {
  const int lane = threadIdx.x;
  const int tb   = blockIdx.x;
  const int half = lane >> 4, l15 = lane & 15, koff = half << 1;
  const float* Arow = ea2 + (size_t)(tb * 16 + l15) * 64;
  __builtin_prefetch(ea2 + (size_t)(tb * 16 + 16 + l15) * 64, 0, 1);

  // A tile (K=64): 16 v2f fragments, loaded once, reused for every N tile
  v2f av[16];
#pragma unroll
  for (int kk = 0; kk < 16; ++kk) av[kk] = *(const v2f*)(Arow + kk * 4 + koff);

  int sr[8], dr[8];
#pragma unroll
  for (int r = 0; r < 8; ++r) {
    const int e = tb * 16 + r + (half << 3);
    sr[r] = src[e];
    dr[r] = dst[e];
  }
  float sacc[8] = {0.f,0.f,0.f,0.f,0.f,0.f,0.f,0.f};

  const int ntiles = HC >> 4;
  for (int nt = 0; nt < ntiles; ++nt) {
    const int nbase = nt << 4;
    const int col   = nbase + l15;
    v8f acc = {0.f,0.f,0.f,0.f,0.f,0.f,0.f,0.f};
#pragma unroll
    for (int kk = 0; kk < 16; ++kk) {
      v2f b;
      b.x = We[(size_t)(kk * 4 + koff)     * HC + col];
      b.y = We[(size_t)(kk * 4 + koff + 1) * HC + col];
      acc = wmma4(av[kk], b, acc);
    }
    const float atv = att[col];  // att is [H,C] flat; col indexes it directly
#pragma unroll
    for (int r = 0; r < 8; ++r) {
      float m = acc[r] + xl[(size_t)sr[r] * HC + col] + xr[(size_t)dr[r] * HC + col];
      m = (m > 0.f) ? m : 0.2f * m;
      float p = m * atv;
      p += __shfl_xor(p, 1);
      p += __shfl_xor(p, 2);
      p += __shfl_xor(p, 4);
      p += __shfl_xor(p, 8);
      sacc[r] += p;
    }
    if (((nbase + 16) % C) == 0) {   // head h complete
      const int h = nbase / C;
#pragma unroll
      for (int r = 0; r < 8; ++r) {
        if (l15 == r) {              // lanes r and 16+r hold rows r and r+8
          const int e = tb * 16 + r + (half << 3);
          sbuf[(size_t)e * 4 + h] = sacc[r];
          atomicMaxF(&emax[dr[r] * 4 + h], sacc[r]);
        }
        sacc[r] = 0.f;
      }
    }
  }
}

// pass B: ex = exp(score - emax[dst]); segment-sum into den
__global__ void gat_expsum_kernel(float* __restrict__ sbuf, const float* __restrict__ emax,
                                  const int* __restrict__ dst, float* __restrict__ den, int E2) {
  int i = blockIdx.x * blockDim.x + threadIdx.x;
  if (i >= E2 * 4) return;
  int e = i >> 2, h = i & 3;
  int d = dst[e];
  float ex = __expf(sbuf[i] - emax[d * 4 + h]);
  sbuf[i] = ex;
  atomicAdd(&den[d * 4 + h], ex);
}

// pass C: alpha = ex/den; scatter out[dst] += xl[src]*alpha; write alpha out
__global__ void gat_aggr_kernel(const float* __restrict__ sbuf, const float* __restrict__ den,
                                const float* __restrict__ xl,
                                const int* __restrict__ src, const int* __restrict__ dst,
                                float* __restrict__ out, float* __restrict__ aout,
                                int HC, int C, int E2) {
  size_t i = (size_t)blockIdx.x * blockDim.x + threadIdx.x;
  if (i >= (size_t)E2 * HC) return;
  int e = (int)(i / HC), j = (int)(i % HC);
  int h = j / C;
  int d = dst[e];
  float al = sbuf[(size_t)e * 4 + h] / (den[d * 4 + h] + 1e-16f);
  if (j == h * C) aout[(size_t)e * 4 + h] = al;
  atomicAdd(&out[(size_t)d * HC + j], xl[(size_t)src[e] * HC + j] * al);
}

// fused (x += gat_bias) -> LayerNorm -> leaky(0.01), in place, one wave/row
__global__ void ln_leaky_kernel(float* __restrict__ x, const float* __restrict__ gatb,
                                const float* __restrict__ g, const float* __restrict__ b,
                                int HC) {
  const int row = blockIdx.x, lane = threadIdx.x;
  const int n = HC >> 5;           // 4 or 8
  float vals[8];
  float sum = 0.f, sq = 0.f;
  for (int i = 0; i < n; ++i) {
    int c = lane + (i << 5);
    float v = x[(size_t)row * HC + c] + gatb[c];
    vals[i] = v;
    sum += v;
    sq  += v * v;
  }
  for (int m = 1; m <= 16; m <<= 1) { sum += __shfl_xor(sum, m); sq += __shfl_xor(sq, m); }
  const float mean = sum / HC;
  const float var  = sq / HC - mean * mean;
  const float rstd = rsqrtf(var + 1e-5f);
  for (int i = 0; i < n; ++i) {
    int c = lane + (i << 5);
    float y = (vals[i] - mean) * rstd * g[c] + b[c];
    x[(size_t)row * HC + c] = (y > 0.f) ? y : 0.01f * y;
  }
}

// node_probs = h2 @ W3 + b3  (K=64, N=1), one wave per row
__global__ void dot64_kernel(const float* __restrict__ h2, const float* __restrict__ W3,
                             const float* __restrict__ b3, float* __restrict__ out) {
  const int row = blockIdx.x, lane = threadIdx.x;
  float s = h2[(size_t)row * 64 + lane] * W3[lane]
          + h2[(size_t)row * 64 + lane + 32] * W3[lane + 32];
  for (int m = 1; m <= 16; m <<= 1) s += __shfl_xor(s, m);
  if (lane == 0) out[row] = s + b3[0];
}

// ---------------------------------------------------------------------------
extern "C" void kernel_launch(void* const* d_in, const int* in_sizes, int n_in,
                              void* d_out, int out_size, void* d_ws, size_t ws_size,
                              hipStream_t stream) {
  constexpr int N  = 20000;
  constexpr int E  = 320000;
  constexpr int E2 = E + N;          // 340000

  int k = 0;
  const float* x_in    = (const float*)d_in[k++];
  const int*   eindex  = (const int*)d_in[k++];
  const float* e_attr  = (const float*)d_in[k++];
  const float* bn_g    = (const float*)d_in[k++];
  const float* bn_b    = (const float*)d_in[k++];
  const float* bn_m    = (const float*)d_in[k++];
  const float* bn_v    = (const float*)d_in[k++];
  const float* ee_W1   = (const float*)d_in[k++];
  const float* ee_b1   = (const float*)d_in[k++];
  const float* ee_W2   = (const float*)d_in[k++];
  const float* ee_b2   = (const float*)d_in[k++];
  const float *gWl[3], *gbl[3], *gWr[3], *gbr[3], *gWe[3], *gatt[3], *gbias[3];
  for (int l = 0; l < 3; ++l) {
    gWl[l]   = (const float*)d_in[k++];
    gbl[l]   = (const float*)d_in[k++];
    gWr[l]   = (const float*)d_in[k++];
    gbr[l]   = (const float*)d_in[k++];
    gWe[l]   = (const float*)d_in[k++];
    gatt[l]  = (const float*)d_in[k++];
    gbias[l] = (const float*)d_in[k++];
  }
  const float* ln_g[3]; const float* ln_b[3];
  for (int l = 0; l < 3; ++l) { ln_g[l] = (const float*)d_in[k++]; ln_b[l] = (const float*)d_in[k++]; }
  const float* skip_W  = (const float*)d_in[k++];
  const float* skip_b  = (const float*)d_in[k++];
  const float* alpha_p = (const float*)d_in[k++];
  const float* fp_W    = (const float*)d_in[k++];
  const float* fp_b    = (const float*)d_in[k++];
  const float* np_W1   = (const float*)d_in[k++];
  const float* np_b1   = (const float*)d_in[k++];
  const float* np_W2   = (const float*)d_in[k++];
  const float* np_b2   = (const float*)d_in[k++];
  const float* np_W3   = (const float*)d_in[k++];
  const float* np_b3   = (const float*)d_in[k++];

  // output regions: xf[20000,256], node_probs[20000,1], a1,a2,a3 [340000,4]
  float* out_f = (float*)d_out;
  float* xf    = out_f;
  float* np_o  = out_f + (size_t)N * 256;
  float* a_o[3] = { np_o + N,
                    np_o + N + (size_t)E2 * 4,
                    np_o + N + (size_t)E2 * 8 };

  // workspace bump allocator
  char* wp = (char*)d_ws;
  auto alloc = [&](size_t bytes) -> void* {
    void* r = (void*)wp;
    wp += (bytes + 255) & ~size_t(255);
    return r;
  };
  float* x_bn  = (float*)alloc((size_t)N * 64 * 4);
  float* ea2   = (float*)alloc((size_t)E2 * 64 * 4);
  int*   src2  = (int*)alloc((size_t)E2 * 4);
  int*   dst2  = (int*)alloc((size_t)E2 * 4);
  float* deg   = (float*)alloc((size_t)N * 4);
  float* xl    = (float*)alloc((size_t)N * 256 * 4);
  float* xr    = (float*)alloc((size_t)N * 256 * 4);
  float* emax  = (float*)alloc((size_t)N * 4 * 4);
  float* den   = (float*)alloc((size_t)N * 4 * 4);
  float* sbuf  = (float*)alloc((size_t)E2 * 4 * 4);
  float* x1    = (float*)alloc((size_t)N * 128 * 4);
  float* skip1 = (float*)alloc((size_t)N * 128 * 4);
  float* x2    = (float*)alloc((size_t)N * 128 * 4);
  float* x3    = (float*)alloc((size_t)N * 256 * 4);
  float* h1    = (float*)alloc((size_t)N * 128 * 4);
  float* h2    = (float*)alloc((size_t)N * 64 * 4);

  const int TB = 256;

  // 1) BatchNorm
  bn_kernel<<<(N * 64 + TB - 1) / TB, TB, 0, stream>>>(x_in, bn_g, bn_b, bn_m, bn_v,
                                                       x_bn, N * 64);
  // 2) fused edge encoder -> ea2[0:E]
  edge_encoder_kernel<<<E / 16, 32, 0, stream>>>(e_attr, ee_W1, ee_b1, ee_W2, ee_b2, ea2);
  // 3) self loops
  build_edges_kernel<<<(E2 + TB - 1) / TB, TB, 0, stream>>>(eindex, src2, dst2, E, N);
  hipMemsetAsync(deg, 0, (size_t)N * 4, stream);
  hipMemsetAsync(ea2 + (size_t)E * 64, 0, (size_t)N * 64 * 4, stream);
  loop_accum_kernel<<<(E * 64 + TB - 1) / TB, TB, 0, stream>>>(ea2, dst2, deg, E);
  loop_div_kernel<<<(N * 64 + TB - 1) / TB, TB, 0, stream>>>(ea2, deg, E, N);

  // 4) GAT layers
  auto run_layer = [&](int l, const float* xin, int din, int HC, int C,
                       float* outx, float* aout) {
    const int mt = N / 16;
    wmma_gemm_kernel<0><<<mt * (HC / 16), 32, 0, stream>>>(
        xin, gWl[l], gbl[l], nullptr, nullptr, 0.f, xl, N, din, HC);
    wmma_gemm_kernel<0><<<mt * (HC / 16), 32, 0, stream>>>(
        xin, gWr[l], gbr[l], nullptr, nullptr, 0.f, xr, N, din, HC);
    fill_kernel<<<(N * 4 + TB - 1) / TB, TB, 0, stream>>>(emax, -3.0e38f, N * 4);
    hipMemsetAsync(den, 0, (size_t)N * 4 * 4, stream);
    hipMemsetAsync(outx, 0, (size_t)N * HC * 4, stream);
    gat_score_kernel<<<E2 / 16, 32, 0, stream>>>(ea2, gWe[l], xl, xr, gatt[l],
                                                 src2, dst2, sbuf, emax, HC, C);
    gat_expsum_kernel<<<(E2 * 4 + TB - 1) / TB, TB, 0, stream>>>(sbuf, emax, dst2, den, E2);
    const size_t tot = (size_t)E2 * HC;
    gat_aggr_kernel<<<(unsigned)((tot + TB - 1) / TB), TB, 0, stream>>>(
        sbuf, den, xl, src2, dst2, outx, aout, HC, C, E2);
    ln_leaky_kernel<<<N, 32, 0, stream>>>(outx, gbias[l], ln_g[l], ln_b[l], HC);
  };

  run_layer(0, x_bn, 64, 128, 32, x1, a_o[0]);
  // skip1 = 0.01*(x_bn@skip_W + skip_b) + x1
  wmma_gemm_kernel<3><<<(N / 16) * (128 / 16), 32, 0, stream>>>(
      x_bn, skip_W, skip_b, x1, nullptr, 0.f, skip1, N, 64, 128);
  run_layer(1, skip1, 128, 128, 32, x2, a_o[1]);
  run_layer(2, x2, 128, 256, 64, x3, a_o[2]);

  // 5) xf = sigmoid(alpha_p)*(x1@fp_W+fp_b) + (1-s)*x3  -> d_out
  wmma_gemm_kernel<4><<<(N / 16) * (256 / 16), 32, 0, stream>>>(
      x1, fp_W, fp_b, x3, alpha_p, 0.f, xf, N, 128, 256);

  // 6) node MLP
  wmma_gemm_kernel<2><<<(N / 16) * (128 / 16), 32, 0, stream>>>(
      xf, np_W1, np_b1, nullptr, nullptr, 0.f, h1, N, 256, 128);
  wmma_gemm_kernel<2><<<(N / 16) * (64 / 16), 32, 0, stream>>>(
      h1, np_W2, np_b2, nullptr, nullptr, 0.f, h2, N, 128, 64);
  dot64_kernel<<<N, 32, 0, stream>>>(h2, np_W3, np_b3, np_o);
}